// PointPillarsScatter_86517821214998
// MI455X (gfx1250) — compile-verified
//
#include <hip/hip_runtime.h>

// PointPillarsScatter for MI455X (gfx1250).
// Inverted scatter->gather:
//   pass1: map[cell] = -1                     (4 MB scratch, L2-resident)
//   pass2: map[b,y,x] = p                     (unique coords, race-free)
//   pass3: out[b,c,y,x] = map>=0 ? feat[p,c] : 0
// Pass 3 writes the 256 MB output with coalesced 128-bit NONTEMPORAL stores
// (th:TH_STORE_NT) so the write stream does not evict the 102 MB feature
// array from the 192 MB L2; features then stream from HBM exactly once.
// Each pass-3 thread covers 4 channels x 4 cells: one int4 map load and four
// 16 B feature gathers per 16 output elements (4x fewer L2 lane-requests
// than a scalar gather), plus an in-register 4x4 transpose.
// Roofline: ~256 MB W + ~108 MB R ~= 365 MB -> ~16 us @ 23.3 TB/s.

typedef __attribute__((ext_vector_type(4))) float v4f;
typedef __attribute__((ext_vector_type(4))) int   v4i;

#define PNY 512
#define PNX 512
#define PC  64
#define PCELLS (PNY * PNX)        // cells per batch image (262144)
#define PLANE4 (PCELLS / 4)       // float4 elements per channel plane (65536)

// ---------------------------------------------------------------------------
// Pass 1: map[cell] = -1 (128-bit regular-temporal stores; map is hot data).
// ---------------------------------------------------------------------------
__global__ void pp_init_map(int* __restrict__ map, int n4) {
    int i = blockIdx.x * blockDim.x + threadIdx.x;
    if (i < n4) {
        v4i m = { -1, -1, -1, -1 };
        ((v4i*)map)[i] = m;
    }
}

// ---------------------------------------------------------------------------
// Pass 2: map[b*NY*NX + y*NX + x] = p. One b128 coalesced coord load/point.
// ---------------------------------------------------------------------------
__global__ void pp_scatter_idx(const int* __restrict__ coords,
                               int* __restrict__ map, int P) {
    int p = blockIdx.x * blockDim.x + threadIdx.x;
    if (p < P) {
        v4i c = ((const v4i*)coords)[p];           // {b, z, y, x}
        int flat = c.x * PCELLS + c.z * PNX + c.w;
        map[flat] = p;
    }
}

// ---------------------------------------------------------------------------
// Pass 3: each thread produces a 4x4 (channel x cell) output micro-tile.
//   i -> (b, cg, y, x4):  x4 = 4-cell group, cg = 4-channel group (c = 4*cg).
//   plane index b*64 + 4*cg + j == bcg*4 + j, with bcg = b*16 + cg.
// ---------------------------------------------------------------------------
__global__ void pp_gather_out(const float* __restrict__ feat,
                              const int* __restrict__ map,
                              float* __restrict__ out, int n16) {
    int i = blockIdx.x * blockDim.x + threadIdx.x;
    if (i >= n16) return;

    int x4  = i & (PNX / 4 - 1);       // 0..127
    int y   = (i >> 7) & (PNY - 1);    // 0..511
    int bcg = i >> 16;                 // b*16 + cg
    int c   = (bcg & 15) << 2;         // channel base 0,4,...,60
    int b   = bcg >> 4;

    // One 128-bit map load covers 4 cells (L2-resident, reused 16x per int4).
    v4i m = ((const v4i*)(map + b * PCELLS + y * PNX))[x4];

    const v4f zero = { 0.0f, 0.0f, 0.0f, 0.0f };

    // Branchless 16 B gathers: clamp empty cells to row 0 (one hot cache
    // line), then select zero. Addresses are 16 B aligned (c % 4 == 0).
    int i0 = ((m.x < 0 ? 0 : m.x) << 6) + c;
    int i1 = ((m.y < 0 ? 0 : m.y) << 6) + c;
    int i2 = ((m.z < 0 ? 0 : m.z) << 6) + c;
    int i3 = ((m.w < 0 ? 0 : m.w) << 6) + c;
    v4f f0 = *(const v4f*)(feat + i0);
    v4f f1 = *(const v4f*)(feat + i1);
    v4f f2 = *(const v4f*)(feat + i2);
    v4f f3 = *(const v4f*)(feat + i3);
    f0 = (m.x < 0) ? zero : f0;
    f1 = (m.y < 0) ? zero : f1;
    f2 = (m.z < 0) ? zero : f2;
    f3 = (m.w < 0) ? zero : f3;

    // In-register 4x4 transpose: channel j across the 4 cells.
    v4f o0, o1, o2, o3;
    o0.x = f0.x; o0.y = f1.x; o0.z = f2.x; o0.w = f3.x;
    o1.x = f0.y; o1.y = f1.y; o1.z = f2.y; o1.w = f3.y;
    o2.x = f0.z; o2.y = f1.z; o2.z = f2.z; o2.w = f3.z;
    o3.x = f0.w; o3.y = f1.w; o3.z = f2.w; o3.w = f3.w;

    // 4 coalesced NT b128 stores, one per channel plane.
    int base4 = (bcg << 2) * PLANE4 + y * (PNX / 4) + x4;
    __builtin_nontemporal_store(o0, (v4f*)out + base4);
    __builtin_nontemporal_store(o1, (v4f*)out + base4 + PLANE4);
    __builtin_nontemporal_store(o2, (v4f*)out + base4 + 2 * PLANE4);
    __builtin_nontemporal_store(o3, (v4f*)out + base4 + 3 * PLANE4);
}

// ---------------------------------------------------------------------------
extern "C" void kernel_launch(void* const* d_in, const int* in_sizes, int n_in,
                              void* d_out, int out_size, void* d_ws, size_t ws_size,
                              hipStream_t stream) {
    const float* feat   = (const float*)d_in[0];   // [P, 64] f32
    const int*   coords = (const int*)d_in[1];     // [P, 4]  i32
    float*       out    = (float*)d_out;           // [B, 64, 512, 512] f32
    int*         map    = (int*)d_ws;              // [B * 512 * 512] i32 scratch

    const int P = in_sizes[1] / 4;
    const int B = out_size / (PC * PNY * PNX);     // derive batch from out size
    const int ncells4 = (B * PCELLS) / 4;          // map in int4 units
    const int n16     = out_size / 16;             // 4x4 micro-tiles

    // Stream order provides pass1 -> pass2 -> pass3 dependencies (graph-safe).
    pp_init_map<<<(ncells4 + 255) / 256, 256, 0, stream>>>(map, ncells4);
    pp_scatter_idx<<<(P + 255) / 256, 256, 0, stream>>>(coords, map, P);
    pp_gather_out<<<(n16 + 255) / 256, 256, 0, stream>>>(feat, map, out, n16);
}